// MultiHeadedAttention_17454747090949
// MI455X (gfx1250) — compile-verified
//
#include <hip/hip_runtime.h>
#include <math.h>

typedef float v2f __attribute__((ext_vector_type(2)));
typedef float v8f __attribute__((ext_vector_type(8)));

#define LN_EPS 1e-5f

// Async DMA: global -> LDS, 16 bytes, tracked by ASYNCcnt (CDNA5).
// LDS operand is the byte offset within the wave's LDS allocation: for the
// generic-pointer aperture scheme, addr[31:0] of a __shared__ pointer IS the
// LDS offset, so a 32-bit truncation is the correct VDST value.
__device__ __forceinline__ void async_ld_b128(const float* gptr, const float* lptr)
{
    unsigned ldso = (unsigned)(uintptr_t)lptr;
    asm volatile("global_load_async_to_lds_b128 %0, %1, off"
                 :: "v"(ldso), "v"(gptr) : "memory");
}
__device__ __forceinline__ void wait_async0()
{
    asm volatile("s_wait_asynccnt 0x0" ::: "memory");
}

// =====================================================================
// Generic fp32 WMMA GEMM: out[M x N] = X[M x K] @ W[K x N] + bias[N]
// Requires: M % 64 == 0, K % 32 == 0.  N arbitrary (guarded).
// Block: 512 threads = 16 waves; each wave owns a 16x16 tile of a 64x64
// macro tile (V_WMMA_F32_16X16X4_F32 — problem is HBM-bound at 23.3 TB/s,
// so full fp32 matrix precision costs nothing).
// K is consumed in 32-deep slabs, double-buffered in LDS and filled by
// GLOBAL_LOAD_ASYNC_TO_LDS_B128 so global traffic overlaps the WMMAs.
// =====================================================================
__global__ __launch_bounds__(512)
void gemm_wmma_f32(const float* __restrict__ X, const float* __restrict__ W,
                   const float* __restrict__ bias, float* __restrict__ out,
                   int M, int K, int N)
{
    // A rows: stride 36 floats (144B: 16B-aligned, banks 36*l%64 distinct)
    // B rows: stride 72 floats (288B: 16B-aligned, half-wave bank sets disjoint)
    __shared__ __align__(16) float As[2][64][36];
    __shared__ __align__(16) float Bs[2][32][72];

    const int mBlock = blockIdx.x * 64;
    const int nBlock = blockIdx.y * 64;
    const int tid    = threadIdx.x;
    const int lane   = tid & 31;
    const int wave   = tid >> 5;            // 0..15
    const int wm     = (wave >> 2) << 4;    // 0,16,32,48
    const int wn     = (wave & 3) << 4;
    const int l16    = lane & 15;
    const int kh     = (lane >> 4) << 1;    // 0 or 2 (K sub-offset per half-wave)

    // staging map: one 16B chunk of A and one of B per thread per slab
    const int rowA = tid >> 3;              // 0..63
    const int colA = (tid & 7) << 2;        // 0..28
    const int rowB = tid >> 4;              // 0..31
    const int colB = (tid & 15) << 2;       // 0..60

    const bool fastN4 = ((nBlock + 64) <= N) && ((N & 3) == 0);

    v8f acc = {0.f,0.f,0.f,0.f,0.f,0.f,0.f,0.f};

    auto stage = [&](int slab, int buf) {
        const int k0 = slab << 5;
        async_ld_b128(X + (size_t)(mBlock + rowA) * K + k0 + colA,
                      &As[buf][rowA][colA]);
        if (fastN4) {
            async_ld_b128(W + (size_t)(k0 + rowB) * N + nBlock + colB,
                          &Bs[buf][rowB][colB]);
        } else {
            #pragma unroll
            for (int i = 0; i < 4; ++i) {
                const int col = nBlock + colB + i;
                Bs[buf][rowB][colB + i] =
                    (col < N) ? W[(size_t)(k0 + rowB) * N + col] : 0.0f;
            }
        }
    };

    stage(0, 0);
    wait_async0();
    __syncthreads();

    const int NSLAB = K >> 5;
    for (int s = 0; s < NSLAB; ++s) {
        const int buf = s & 1;
        if (s + 1 < NSLAB) stage(s + 1, buf ^ 1);   // overlap with compute

        #pragma unroll
        for (int kk = 0; kk < 32; kk += 4) {
            v2f a, bf;
            a.x  = As[buf][wm + l16][kk + kh];
            a.y  = As[buf][wm + l16][kk + kh + 1];
            bf.x = Bs[buf][kk + kh][wn + l16];
            bf.y = Bs[buf][kk + kh + 1][wn + l16];
            acc = __builtin_amdgcn_wmma_f32_16x16x4_f32(
                      false, a, false, bf, (short)0, acc, false, false);
        }
        wait_async0();        // next slab's LDS writes landed
        __syncthreads();      // all waves done reading current slab
    }

    // epilogue: bias + store (C layout: M = v + 8*(lane/16), N = lane%16)
    const int col = nBlock + wn + l16;
    if (col < N) {
        const float bv = bias[col];
        const int rowBase = mBlock + wm + ((lane >> 4) << 3);
        #pragma unroll
        for (int v = 0; v < 8; ++v) {
            out[(size_t)(rowBase + v) * N + col] = acc[v] + bv;
        }
    }
}

// =====================================================================
// Row LayerNorm + ReLU, in place. One block (256 thr) per row, width w<=256.
// =====================================================================
__global__ __launch_bounds__(256)
void ln_relu_kernel(float* __restrict__ buf, const float* __restrict__ g,
                    const float* __restrict__ b, int w)
{
    __shared__ float red[256];
    const int row = blockIdx.x;
    const int tid = threadIdx.x;
    const float v = (tid < w) ? buf[(size_t)row * w + tid] : 0.0f;

    red[tid] = v;
    __syncthreads();
    #pragma unroll
    for (int s = 128; s > 0; s >>= 1) {
        if (tid < s) red[tid] += red[tid + s];
        __syncthreads();
    }
    const float mean = red[0] / (float)w;
    __syncthreads();

    red[tid] = v * v;
    __syncthreads();
    #pragma unroll
    for (int s = 128; s > 0; s >>= 1) {
        if (tid < s) red[tid] += red[tid + s];
        __syncthreads();
    }
    const float var = red[0] / (float)w - mean * mean;

    if (tid < w) {
        const float y = (v - mean) * rsqrtf(var + LN_EPS) * g[tid] + b[tid];
        buf[(size_t)row * w + tid] = fmaxf(y, 0.0f);
    }
}

// =====================================================================
// Attention: grid = B blocks, 256 threads = 8 waves, wave h = head h.
// Lane m (<17) owns K-row m; per-row softmax via wave32 shfl_xor.
// Skeleton mask from constant bitmasks + sigmoid(relation)*dyn_scale.
// Writes attention output into X (may alias Q: each wave reads only its
// own [b, :, h*64..h*64+63] slice strictly before writing it).
// =====================================================================
__constant__ unsigned c_dmask[17] = {
    0x92u, 0x4u, 0x8u, 0u, 0x20u, 0x40u, 0u, 0x100u, 0x4A00u,
    0x400u, 0u, 0x1000u, 0x2000u, 0u, 0x8000u, 0x10000u, 0u };
__constant__ unsigned c_smask[17] = {
    0u, 0x10u, 0x20u, 0x40u, 0x2u, 0x4u, 0x8u, 0u, 0u,
    0u, 0u, 0x4000u, 0x8000u, 0x10000u, 0x800u, 0x1000u, 0x2000u };

__global__ __launch_bounds__(256)
void attn_kernel(const float* __restrict__ Q, const float* __restrict__ Kb,
                 const float* __restrict__ Vb, const float* __restrict__ rel,
                 const float* __restrict__ p_dw, const float* __restrict__ p_sw,
                 const float* __restrict__ p_selfw, const float* __restrict__ p_dsc,
                 float* __restrict__ X)
{
    __shared__ float P[8][17][20];
    const int b    = blockIdx.x;
    const int h    = threadIdx.x >> 5;
    const int lane = threadIdx.x & 31;
    const float dw = *p_dw, sw = *p_sw, sfw = *p_selfw, dsc = *p_dsc;
    const size_t rowBase = (size_t)b * 17;

    // lane m holds K[b, m, h*64 .. +63]
    float kreg[64];
    const int krow = (lane < 17) ? lane : 0;
    const float* Kp = Kb + (rowBase + krow) * 512 + h * 64;
    #pragma unroll
    for (int d = 0; d < 64; ++d) kreg[d] = (lane < 17) ? Kp[d] : 0.0f;

    for (int n = 0; n < 17; ++n) {
        const float* Qp = Q + (rowBase + n) * 512 + h * 64;
        float acc = 0.0f;
        #pragma unroll
        for (int d = 0; d < 64; ++d) acc = fmaf(Qp[d], kreg[d], acc);

        float sc = -__builtin_inff();
        if (lane < 17) {
            const float dyn = dsc / (1.0f + __expf(-rel[(rowBase + n) * 17 + lane]));
            const float mask = 1.0f
                + dw * (float)((c_dmask[n] >> lane) & 1u)
                + sw * (float)((c_smask[n] >> lane) & 1u)
                + ((n == lane) ? sfw : 0.0f)
                + dyn;
            sc = acc * 0.125f * mask;          // 1/sqrt(64)
        }
        float mx = sc;
        #pragma unroll
        for (int off = 16; off > 0; off >>= 1)
            mx = fmaxf(mx, __shfl_xor(mx, off, 32));
        const float e = (lane < 17) ? __expf(sc - mx) : 0.0f;
        float se = e;
        #pragma unroll
        for (int off = 16; off > 0; off >>= 1)
            se += __shfl_xor(se, off, 32);
        if (lane < 17) P[h][n][lane] = e / se;
    }

    // lane holds V columns d = lane and lane+32 for all 17 rows
    float vreg[34];
    #pragma unroll
    for (int m = 0; m < 17; ++m) {
        const float* Vp = Vb + (rowBase + m) * 512 + h * 64;
        vreg[2 * m]     = Vp[lane];
        vreg[2 * m + 1] = Vp[lane + 32];
    }
    for (int n = 0; n < 17; ++n) {
        float a0 = 0.0f, a1 = 0.0f;
        #pragma unroll
        for (int m = 0; m < 17; ++m) {
            const float p = P[h][n][m];
            a0 = fmaf(p, vreg[2 * m], a0);
            a1 = fmaf(p, vreg[2 * m + 1], a1);
        }
        float* xp = X + (rowBase + n) * 512 + h * 64;
        xp[lane]      = a0;
        xp[lane + 32] = a1;
    }
}

// =====================================================================
// Launch pipeline (all on `stream`, graph-capture safe, ws-only scratch).
// =====================================================================
extern "C" void kernel_launch(void* const* d_in, const int* in_sizes, int n_in,
                              void* d_out, int out_size, void* d_ws, size_t ws_size,
                              hipStream_t stream)
{
    (void)in_sizes; (void)n_in; (void)out_size; (void)ws_size;

    const float* query = (const float*)d_in[0];
    const float* key   = (const float*)d_in[1];
    const float* value = (const float*)d_in[2];
    const float* Wq  = (const float*)d_in[3];   const float* bq  = (const float*)d_in[4];
    const float* Wk  = (const float*)d_in[5];   const float* bk  = (const float*)d_in[6];
    const float* Wv  = (const float*)d_in[7];   const float* bv  = (const float*)d_in[8];
    const float* Wo  = (const float*)d_in[9];   const float* bo  = (const float*)d_in[10];
    const float* A1W = (const float*)d_in[11];  const float* A1b = (const float*)d_in[12];
    const float* g1  = (const float*)d_in[13];  const float* b1  = (const float*)d_in[14];
    const float* A2W = (const float*)d_in[15];  const float* A2b = (const float*)d_in[16];
    const float* g2  = (const float*)d_in[17];  const float* b2  = (const float*)d_in[18];
    const float* Wp  = (const float*)d_in[19];  const float* bp  = (const float*)d_in[20];
    const float* dwp = (const float*)d_in[21];  const float* swp = (const float*)d_in[22];
    const float* sfw = (const float*)d_in[23];  const float* dsc = (const float*)d_in[24];

    const int Mrows = 4096 * 17;                 // 69632, divisible by 64

    float* ws  = (float*)d_ws;
    float* Qb  = ws;                              // (M,512) — later reused as attn output X
    float* Kb  = Qb + (size_t)Mrows * 512;        // (M,512)
    float* Vbf = Kb + (size_t)Mrows * 512;        // (M,512)
    float* h1  = Vbf + (size_t)Mrows * 512;       // (M,256)
    float* h2  = h1 + (size_t)Mrows * 256;        // (M,128)
    float* rel = h2 + (size_t)Mrows * 128;        // (M,17)

    const dim3 blk(512);
    const dim3 g512(Mrows / 64, 8);               // N=512
    const dim3 g256(Mrows / 64, 4);               // N=256
    const dim3 g128(Mrows / 64, 2);               // N=128
    const dim3 g17 (Mrows / 64, 1);               // N=17

    // Q/K/V projections
    gemm_wmma_f32<<<g512, blk, 0, stream>>>(query, Wq, bq, Qb,  Mrows, 512, 512);
    gemm_wmma_f32<<<g512, blk, 0, stream>>>(key,   Wk, bk, Kb,  Mrows, 512, 512);
    gemm_wmma_f32<<<g512, blk, 0, stream>>>(value, Wv, bv, Vbf, Mrows, 512, 512);

    // dynamic-mask MLP: 512->256 (LN,relu) ->128 (LN,relu) ->17 (sigmoid fused in attn)
    gemm_wmma_f32<<<g256, blk, 0, stream>>>(query, A1W, A1b, h1, Mrows, 512, 256);
    ln_relu_kernel<<<Mrows, 256, 0, stream>>>(h1, g1, b1, 256);
    gemm_wmma_f32<<<g128, blk, 0, stream>>>(h1, A2W, A2b, h2, Mrows, 256, 128);
    ln_relu_kernel<<<Mrows, 256, 0, stream>>>(h2, g2, b2, 128);
    gemm_wmma_f32<<<g17,  blk, 0, stream>>>(h2, Wp, bp, rel, Mrows, 128, 17);

    // attention (writes X into Qb — per-wave read-before-write on disjoint slices)
    attn_kernel<<<4096, 256, 0, stream>>>(Qb, Kb, Vbf, rel, dwp, swp, sfw, dsc, Qb);

    // output projection
    gemm_wmma_f32<<<g512, blk, 0, stream>>>(Qb, Wo, bo, (float*)d_out, Mrows, 512, 512);
}